// Conv_agg_15736760172734
// MI455X (gfx1250) — compile-verified
//
#include <hip/hip_runtime.h>
#include <hip/hip_bf16.h>

// ---------------------------------------------------------------------------
// Problem constants (from the reference)
// ---------------------------------------------------------------------------
#define N_NODES 10000
#define N_EDGES 320000
#define K_HOPS  3
#define IN_CH   128
#define OUT_CH  128

typedef __attribute__((ext_vector_type(2))) float v2f;
typedef __attribute__((ext_vector_type(8))) float v8f;

// ---------------------------------------------------------------------------
// Kernel 1: zero the agg workspace [N_NODES, K_HOPS, IN_CH] (f32)
// ---------------------------------------------------------------------------
__global__ void zero_ws_kernel(float* __restrict__ p, int n) {
    int i = blockIdx.x * blockDim.x + threadIdx.x;
    if (i < n) p[i] = 0.0f;
}

// ---------------------------------------------------------------------------
// Kernel 2: sparse scatter  agg[src, k, :] += X[e,k] * h[dst, :]
// One wave32 per edge; each lane owns 4 contiguous channels (float4 of h).
// 12 f32 atomics per lane -> global_atomic_add_f32 at L2.
// ---------------------------------------------------------------------------
__global__ void scatter_kernel(const float* __restrict__ h,
                               const float* __restrict__ X,
                               const long long* __restrict__ edge_index,
                               float* __restrict__ agg) {
    int wid  = (blockIdx.x * blockDim.x + threadIdx.x) >> 5;  // edge id
    int lane = threadIdx.x & 31;
    if (wid >= N_EDGES) return;  // wave-uniform

    long long src = edge_index[wid];            // row 0 of edge_index
    long long dst = edge_index[N_EDGES + wid];  // row 1 of edge_index

    float x0 = X[wid * 3 + 0];
    float x1 = X[wid * 3 + 1];
    float x2 = X[wid * 3 + 2];

    // coalesced: wave reads 128 contiguous floats of h[dst]
    const float4* hv = reinterpret_cast<const float4*>(h + (size_t)dst * IN_CH);
    float4 hh = hv[lane];

    float* base = agg + (size_t)src * (K_HOPS * IN_CH);
    int ch = lane * 4;

    atomicAdd(base + 0 * IN_CH + ch + 0, x0 * hh.x);
    atomicAdd(base + 0 * IN_CH + ch + 1, x0 * hh.y);
    atomicAdd(base + 0 * IN_CH + ch + 2, x0 * hh.z);
    atomicAdd(base + 0 * IN_CH + ch + 3, x0 * hh.w);

    atomicAdd(base + 1 * IN_CH + ch + 0, x1 * hh.x);
    atomicAdd(base + 1 * IN_CH + ch + 1, x1 * hh.y);
    atomicAdd(base + 1 * IN_CH + ch + 2, x1 * hh.z);
    atomicAdd(base + 1 * IN_CH + ch + 3, x1 * hh.w);

    atomicAdd(base + 2 * IN_CH + ch + 0, x2 * hh.x);
    atomicAdd(base + 2 * IN_CH + ch + 1, x2 * hh.y);
    atomicAdd(base + 2 * IN_CH + ch + 2, x2 * hh.z);
    atomicAdd(base + 2 * IN_CH + ch + 3, x2 * hh.w);
}

// ---------------------------------------------------------------------------
// Kernel 3: fused  out[m, n] = max_k ( agg[m, k, :] @ W[k] )[n] + bias[n]
// One wave per 16x16 output tile, V_WMMA_F32_16X16X4_F32, K=128 in 32 steps.
//
// VGPR layouts (ISA 7.12.2, 32-bit A 16x4 / 32-bit C-D 16x16):
//   A:  lane L: M = L%16, holds K-pair (2*(L/16), 2*(L/16)+1) in v0,v1
//   B:  lane L: N = L%16, holds K-pair (2*(L/16), 2*(L/16)+1) in v0,v1
//   C/D: vgpr r, lane L -> M = r + 8*(L/16), N = L%16
// ---------------------------------------------------------------------------
__global__ void gemm_max_bias_kernel(const float* __restrict__ agg,
                                     const float* __restrict__ weight,
                                     const float* __restrict__ bias,
                                     float* __restrict__ out) {
    const int M_TILES = N_NODES / 16;  // 625
    const int N_TILES = OUT_CH / 16;   // 8

    int gwid = (blockIdx.x * blockDim.x + threadIdx.x) >> 5;
    int lane = threadIdx.x & 31;
    if (gwid >= M_TILES * N_TILES) return;  // wave-uniform (exact grid anyway)

    int mt = gwid / N_TILES;
    int nt = gwid % N_TILES;
    int m_base = mt * 16;
    int n_base = nt * 16;

    int lm = lane & 15;   // M (for A) / N (for B,D) within tile
    int lh = lane >> 4;   // which half of the wave -> which K-pair

    v8f maxv;
#pragma unroll
    for (int r = 0; r < 8; ++r) maxv[r] = -__builtin_inff();

#pragma unroll
    for (int k = 0; k < K_HOPS; ++k) {
        v8f acc = {};
        // A row pointer for this lane: agg[m_base+lm][k][lh*2 + ks]
        const float* Abase =
            agg + ((size_t)(m_base + lm) * K_HOPS + k) * IN_CH + lh * 2;
        // B column pointer for this lane: weight[k][ks + lh*2][n_base+lm]
        const float* Bbase =
            weight + (size_t)k * IN_CH * OUT_CH + n_base + lm;

#pragma unroll
        for (int ks = 0; ks < IN_CH; ks += 4) {
            v2f a = *reinterpret_cast<const v2f*>(Abase + ks);  // 8B aligned
            v2f b;
            b.x = Bbase[(ks + lh * 2 + 0) * OUT_CH];
            b.y = Bbase[(ks + lh * 2 + 1) * OUT_CH];
            // D = A(16x4) x B(4x16) + C, full f32
            acc = __builtin_amdgcn_wmma_f32_16x16x4_f32(
                /*neg_a=*/false, a, /*neg_b=*/false, b,
                /*c_mod=*/(short)0, acc, /*reuse_a=*/false, /*reuse_b=*/false);
        }
#pragma unroll
        for (int r = 0; r < 8; ++r) maxv[r] = fmaxf(maxv[r], acc[r]);
    }

    float bv = bias[n_base + lm];
#pragma unroll
    for (int r = 0; r < 8; ++r) {
        int m = m_base + r + 8 * lh;
        out[(size_t)m * OUT_CH + n_base + lm] = maxv[r] + bv;
    }
}

// ---------------------------------------------------------------------------
// Launch
// ---------------------------------------------------------------------------
extern "C" void kernel_launch(void* const* d_in, const int* in_sizes, int n_in,
                              void* d_out, int out_size, void* d_ws, size_t ws_size,
                              hipStream_t stream) {
    const float*     h    = (const float*)d_in[0];      // [N_NODES, IN_CH]
    const float*     X    = (const float*)d_in[1];      // [N_EDGES, K_HOPS]
    const long long* ei   = (const long long*)d_in[2];  // [2, N_EDGES] int64
    // d_in[3] = batch_node (unused by the computation)
    const float*     W    = (const float*)d_in[4];      // [K_HOPS, IN_CH, OUT_CH]
    const float*     bias = (const float*)d_in[5];      // [OUT_CH]
    float*           out  = (float*)d_out;              // [N_NODES, OUT_CH]
    float*           agg  = (float*)d_ws;               // [N_NODES, K_HOPS, IN_CH]

    const int agg_elems = N_NODES * K_HOPS * IN_CH;  // 3,840,000 floats

    // 1) zero workspace
    zero_ws_kernel<<<(agg_elems + 255) / 256, 256, 0, stream>>>(agg, agg_elems);

    // 2) edge scatter: one wave32 per edge -> E*32 threads
    scatter_kernel<<<(N_EDGES * 32) / 256, 256, 0, stream>>>(h, X, ei, agg);

    // 3) fused WMMA GEMM + max-over-hops + bias: one wave per 16x16 tile
    const int total_waves = (N_NODES / 16) * (OUT_CH / 16);  // 5000
    gemm_max_bias_kernel<<<total_waves / 8, 256, 0, stream>>>(agg, W, bias, out);
}